// SVT_channel_mixing_13941463842884
// MI455X (gfx1250) — compile-verified
//
#include <hip/hip_runtime.h>

// ---------------------------------------------------------------------------
// DTCWT J1 + complex block-MLP channel mixing + inverse DTCWT  (all f32)
// MI455X / gfx1250: wave32, WMMA f32 16x16x4 for the 16x16 block matmuls.
// ---------------------------------------------------------------------------

#define HSZ   56
#define WSZ   56
#define NPIX  (HSZ*WSZ)            // 3136
#define CDIM  64
#define BATCH 32
#define NBLK  4
#define BSZ   16
#define HH    28
#define WH    28
#define NPOS  (BATCH*6*HH*WH)      // 150528 positions per block index
#define NTILE (NBLK*(NPOS/16))     // 37632 WMMA tiles
#define SQ2I  0.70710678118654752440f

typedef float v2f __attribute__((ext_vector_type(2)));
typedef float v8f __attribute__((ext_vector_type(8)));

__constant__ float H0c[13] = {
  -0.0017578125f, 0.0f, 0.022265625f, -0.046875f, -0.0482421875f, 0.296875f,
   0.55546875f, 0.296875f, -0.0482421875f, -0.046875f, 0.022265625f, 0.0f,
  -0.0017578125f };
__constant__ float H1c[19] = {
  -7.0626e-05f, 0.0f, 0.00134189f, -0.00188341f, -0.0071566f, 0.023856f,
   0.0556431f, -0.0516881f, -0.299758f, 0.559431f, -0.299758f, -0.0516881f,
   0.0556431f, 0.023856f, -0.0071566f, -0.00188341f, 0.00134189f, 0.0f,
  -7.0626e-05f };
// G0 = alt(H1) (19 taps), G1 = alt(H0) (13 taps)
__constant__ float G0c[19] = {
   7.0626e-05f, 0.0f, -0.00134189f, -0.00188341f, 0.0071566f, 0.023856f,
  -0.0556431f, -0.0516881f, 0.299758f, 0.559431f, 0.299758f, -0.0516881f,
  -0.0556431f, 0.023856f, 0.0071566f, -0.00188341f, -0.00134189f, 0.0f,
   7.0626e-05f };
__constant__ float G1c[13] = {
  -0.0017578125f, 0.0f, 0.022265625f, 0.046875f, -0.0482421875f, -0.296875f,
   0.55546875f, -0.296875f, -0.0482421875f, 0.046875f, 0.022265625f, 0.0f,
  -0.0017578125f };

__device__ __forceinline__ int symi(int i, int n) {
  if (i < 0)  return -1 - i;
  if (i >= n) return 2*n - 1 - i;
  return i;
}

// column-filter a 2x2 quad at subband pixel (i,j); applies 1/sqrt(2)
__device__ __forceinline__ void colquad(const float* src, const float* f,
                                        int nt, int ctr, int i, int j,
                                        float& a, float& bb, float& cc, float& d)
{
  float y00=0.f, y01=0.f, y10=0.f, y11=0.f;
  int c0 = 2*j, c1 = 2*j + 1;
  for (int t = 0; t < nt; ++t) {
    float ft = f[t];
    int r0 = symi(2*i     + t - ctr, HSZ);
    int r1 = symi(2*i + 1 + t - ctr, HSZ);
    y00 += ft * src[r0*WSZ + c0];  y01 += ft * src[r0*WSZ + c1];
    y10 += ft * src[r1*WSZ + c0];  y11 += ft * src[r1*WSZ + c1];
  }
  a = y00*SQ2I; bb = y01*SQ2I; cc = y10*SQ2I; d = y11*SQ2I;
}

// ---------------------------------------------------------------------------
// Kernel 1: forward DTCWT (one workgroup per (batch, channel) image)
// ---------------------------------------------------------------------------
__global__ __launch_bounds__(256)
void fwd_dtcwt(const float* __restrict__ x, const float* __restrict__ w_ll,
               float* __restrict__ llout, float* __restrict__ Xr,
               float* __restrict__ Xi)
{
  __shared__ float img[NPIX];
  __shared__ float rlo[NPIX];
  __shared__ float rhi[NPIX];
  const int b = blockIdx.x >> 6, c = blockIdx.x & 63;
  const int tid = threadIdx.x;

  const float* xp = x + (size_t)b * NPIX * CDIM + c;
  for (int idx = tid; idx < NPIX; idx += 256) {
    __builtin_prefetch(&xp[(size_t)(idx + 512) * CDIM], 0, 1);
    img[idx] = xp[(size_t)idx * CDIM];
  }
  __syncthreads();

  // row filters (along w)
  for (int idx = tid; idx < NPIX; idx += 256) {
    int h = idx / WSZ, w = idx - h*WSZ;
    float lo = 0.f, hi = 0.f;
#pragma unroll
    for (int t = 0; t < 13; ++t) lo += H0c[t] * img[h*WSZ + symi(w + t - 6, WSZ)];
#pragma unroll
    for (int t = 0; t < 19; ++t) hi += H1c[t] * img[h*WSZ + symi(w + t - 9, WSZ)];
    rlo[idx] = lo; rhi[idx] = hi;
  }
  __syncthreads();

  // LL = colfilt(rlo, H0), scaled by w_ll
  const float* wl  = w_ll + (size_t)c * NPIX;
  float*       llo = llout + (size_t)(b*CDIM + c) * NPIX;
  for (int idx = tid; idx < NPIX; idx += 256) {
    int h = idx / WSZ, w = idx - h*WSZ;
    float a = 0.f;
#pragma unroll
    for (int t = 0; t < 13; ++t) a += H0c[t] * rlo[symi(h + t - 6, HSZ)*WSZ + w];
    llo[idx] = a * wl[idx];
  }

  // subbands -> packed [nb][pos][16] real/imag
  const int nb = c >> 4, k = c & 15;
  for (int p = tid; p < HH*WH; p += 256) {
    int i = p / WH, j = p - i*WH;
    float a, bb, cc, d;
    size_t g0 = ((size_t)nb*NPOS + (size_t)(((b*6 + 0)*HH + i)*WH + j))*BSZ + k;
    size_t g1 = ((size_t)nb*NPOS + (size_t)(((b*6 + 1)*HH + i)*WH + j))*BSZ + k;
    size_t g2 = ((size_t)nb*NPOS + (size_t)(((b*6 + 2)*HH + i)*WH + j))*BSZ + k;
    size_t g3 = ((size_t)nb*NPOS + (size_t)(((b*6 + 3)*HH + i)*WH + j))*BSZ + k;
    size_t g4 = ((size_t)nb*NPOS + (size_t)(((b*6 + 4)*HH + i)*WH + j))*BSZ + k;
    size_t g5 = ((size_t)nb*NPOS + (size_t)(((b*6 + 5)*HH + i)*WH + j))*BSZ + k;
    // lohi = colfilt(rlo, H1) -> bands 15 (s=0) and 165 (s=5)
    colquad(rlo, H1c, 19, 9, i, j, a, bb, cc, d);
    Xr[g0] = a - d;  Xi[g0] = bb + cc;
    Xr[g5] = a + d;  Xi[g5] = bb - cc;
    // hilo = colfilt(rhi, H0) -> bands 75 (s=2) and 105 (s=3)
    colquad(rhi, H0c, 13, 6, i, j, a, bb, cc, d);
    Xr[g2] = a - d;  Xi[g2] = bb + cc;
    Xr[g3] = a + d;  Xi[g3] = bb - cc;
    // hihi = colfilt(rhi, H1) -> bands 45 (s=1) and 135 (s=4)
    colquad(rhi, H1c, 19, 9, i, j, a, bb, cc, d);
    Xr[g1] = a - d;  Xi[g1] = bb + cc;
    Xr[g4] = a + d;  Xi[g4] = bb - cc;
  }
}

// ---------------------------------------------------------------------------
// Kernel 2: complex 2-layer block MLP via WMMA f32 16x16x4 (one wave = one
// 16-position x 16-channel tile; in-place update of Xr/Xi)
// ---------------------------------------------------------------------------
__global__ __launch_bounds__(256)
void mix_wmma(const float* __restrict__ w1, const float* __restrict__ w2,
              const float* __restrict__ b1, const float* __restrict__ b2,
              float* __restrict__ Xr, float* __restrict__ Xi)
{
  __shared__ float smem[8][2*256];          // per-wave staging tiles
  const int tid   = threadIdx.x;
  const int wave  = tid >> 5;
  const int lane  = tid & 31;
  const int half  = lane >> 4;
  const int n     = lane & 15;              // output channel / A row M

  const int tileIdx = blockIdx.x * 8 + wave;          // < NTILE exactly
  const int nb      = tileIdx / (NPOS/16);
  const int posbase = (tileIdx - nb*(NPOS/16)) * 16;

  const float* W1r = w1 + nb*256;
  const float* W1i = w1 + 1024 + nb*256;
  const float* W2r = w2 + nb*256;
  const float* W2i = w2 + 1024 + nb*256;

  // B fragments: chunk kk covers K = 4kk..4kk+3.
  // lane half 0 -> rows {4kk, 4kk+1}, half 1 -> rows {4kk+2, 4kk+3}
  v2f bw1r[4], bw1i[4], bn1i[4], bw2r[4], bw2i[4], bn2i[4];
#pragma unroll
  for (int kk = 0; kk < 4; ++kk) {
    int r0 = 4*kk + 2*half;
    bw1r[kk][0] = W1r[r0*16 + n];      bw1r[kk][1] = W1r[(r0+1)*16 + n];
    bw1i[kk][0] = W1i[r0*16 + n];      bw1i[kk][1] = W1i[(r0+1)*16 + n];
    bw2r[kk][0] = W2r[r0*16 + n];      bw2r[kk][1] = W2r[(r0+1)*16 + n];
    bw2i[kk][0] = W2i[r0*16 + n];      bw2i[kk][1] = W2i[(r0+1)*16 + n];
    bn1i[kk][0] = -bw1i[kk][0];        bn1i[kk][1] = -bw1i[kk][1];
    bn2i[kk][0] = -bw2i[kk][0];        bn2i[kk][1] = -bw2i[kk][1];
  }

  // A fragments: lane's row M = n, K pair = 4kk + 2*half
  const float* xr_base = Xr + ((size_t)nb*NPOS + posbase) * BSZ;
  const float* xi_base = Xi + ((size_t)nb*NPOS + posbase) * BSZ;
  v2f ar[4], ai[4];
#pragma unroll
  for (int kk = 0; kk < 4; ++kk) {
    int ks = 4*kk + 2*half;
    ar[kk] = *reinterpret_cast<const v2f*>(xr_base + n*16 + ks);
    ai[kk] = *reinterpret_cast<const v2f*>(xi_base + n*16 + ks);
  }

  const float b1r_s = b1[      nb*16 + n];
  const float b1i_s = b1[64  + nb*16 + n];
  const float b2r_s = b2[      nb*16 + n];
  const float b2i_s = b2[64  + nb*16 + n];

  v8f r1, i1;
#pragma unroll
  for (int q = 0; q < 8; ++q) { r1[q] = b1r_s; i1[q] = b1i_s; }

  // layer 1: r1 += Xr@W1r + Xi@(-W1i);  i1 += Xr@W1i + Xi@W1r
#pragma unroll
  for (int kk = 0; kk < 4; ++kk) {
    r1 = __builtin_amdgcn_wmma_f32_16x16x4_f32(false, ar[kk], false, bw1r[kk],
                                               (short)0, r1, false, false);
    r1 = __builtin_amdgcn_wmma_f32_16x16x4_f32(false, ai[kk], false, bn1i[kk],
                                               (short)0, r1, false, false);
    i1 = __builtin_amdgcn_wmma_f32_16x16x4_f32(false, ar[kk], false, bw1i[kk],
                                               (short)0, i1, false, false);
    i1 = __builtin_amdgcn_wmma_f32_16x16x4_f32(false, ai[kk], false, bw1r[kk],
                                               (short)0, i1, false, false);
  }
#pragma unroll
  for (int q = 0; q < 8; ++q) { r1[q] = fmaxf(r1[q], 0.f); i1[q] = fmaxf(i1[q], 0.f); }

  // redistribute C/D layout -> A layout through wave-private LDS
  float* ldsR = &smem[wave][0];
  float* ldsI = &smem[wave][256];
#pragma unroll
  for (int v = 0; v < 8; ++v) {
    int m = v + 8*half;
    ldsR[m*16 + n] = r1[v];
    ldsI[m*16 + n] = i1[v];
  }
  v2f a2r[4], a2i[4];
#pragma unroll
  for (int kk = 0; kk < 4; ++kk) {
    int ks = 4*kk + 2*half;
    a2r[kk] = *reinterpret_cast<const v2f*>(ldsR + n*16 + ks);
    a2i[kk] = *reinterpret_cast<const v2f*>(ldsI + n*16 + ks);
  }

  v8f r2, i2;
#pragma unroll
  for (int q = 0; q < 8; ++q) { r2[q] = b2r_s; i2[q] = b2i_s; }

  // layer 2: r2 += R1@W2r + I1@(-W2i);  i2 += R1@W2i + I1@W2r
#pragma unroll
  for (int kk = 0; kk < 4; ++kk) {
    r2 = __builtin_amdgcn_wmma_f32_16x16x4_f32(false, a2r[kk], false, bw2r[kk],
                                               (short)0, r2, false, false);
    r2 = __builtin_amdgcn_wmma_f32_16x16x4_f32(false, a2i[kk], false, bn2i[kk],
                                               (short)0, r2, false, false);
    i2 = __builtin_amdgcn_wmma_f32_16x16x4_f32(false, a2r[kk], false, bw2i[kk],
                                               (short)0, i2, false, false);
    i2 = __builtin_amdgcn_wmma_f32_16x16x4_f32(false, a2i[kk], false, bw2r[kk],
                                               (short)0, i2, false, false);
  }

  // store (in-place over the tile this wave already consumed)
#pragma unroll
  for (int v = 0; v < 8; ++v) {
    int m = v + 8*half;
    size_t g = ((size_t)nb*NPOS + posbase + m) * BSZ + n;
    Xr[g] = r2[v];
    Xi[g] = i2[v];
  }
}

// ---------------------------------------------------------------------------
// Kernel 3: inverse DTCWT (one workgroup per (batch, channel))
// ---------------------------------------------------------------------------
__device__ __forceinline__ float c2q_elem(const float* __restrict__ Xr,
                                          const float* __restrict__ Xi,
                                          size_t ga, size_t gb, int ph, int pw)
{
  if (ph == 0) return (pw == 0) ? SQ2I*(Xr[ga] + Xr[gb]) : SQ2I*(Xi[ga] + Xi[gb]);
  else         return (pw == 0) ? SQ2I*(Xi[ga] - Xi[gb]) : SQ2I*(Xr[gb] - Xr[ga]);
}

__global__ __launch_bounds__(256)
void inv_dtcwt(const float* __restrict__ llin, const float* __restrict__ Xr,
               const float* __restrict__ Xi, float* __restrict__ out)
{
  __shared__ float P0[NPIX];   // hl, then ll
  __shared__ float P1[NPIX];   // hh, then lh
  __shared__ float P2[NPIX];   // hi image
  __shared__ float P3[NPIX];   // lo image
  const int b = blockIdx.x >> 6, c = blockIdx.x & 63;
  const int tid = threadIdx.x;
  const int nb = c >> 4, k = c & 15;

  // stage 1: hl = c2q(band2, band4? no: bands 2,3); hh = c2q(bands 1,4)
  for (int idx = tid; idx < NPIX; idx += 256) {
    int h = idx / WSZ, w = idx - h*WSZ;
    int i = h >> 1, j = w >> 1, ph = h & 1, pw = w & 1;
    size_t g1 = ((size_t)nb*NPOS + (size_t)(((b*6 + 1)*HH + i)*WH + j))*BSZ + k;
    size_t g2 = ((size_t)nb*NPOS + (size_t)(((b*6 + 2)*HH + i)*WH + j))*BSZ + k;
    size_t g3 = ((size_t)nb*NPOS + (size_t)(((b*6 + 3)*HH + i)*WH + j))*BSZ + k;
    size_t g4 = ((size_t)nb*NPOS + (size_t)(((b*6 + 4)*HH + i)*WH + j))*BSZ + k;
    P0[idx] = c2q_elem(Xr, Xi, g2, g3, ph, pw);   // hl
    P1[idx] = c2q_elem(Xr, Xi, g1, g4, ph, pw);   // hh
  }
  __syncthreads();

  // stage 2: hi = colfilt(hl, G0) + colfilt(hh, G1)
  for (int idx = tid; idx < NPIX; idx += 256) {
    int h = idx / WSZ, w = idx - h*WSZ;
    float acc = 0.f;
#pragma unroll
    for (int t = 0; t < 19; ++t) acc += G0c[t] * P0[symi(h + t - 9, HSZ)*WSZ + w];
#pragma unroll
    for (int t = 0; t < 13; ++t) acc += G1c[t] * P1[symi(h + t - 6, HSZ)*WSZ + w];
    P2[idx] = acc;
  }
  __syncthreads();

  // stage 3: reload P0 = scaled LL, P1 = lh = c2q(bands 0,5)
  const float* llo = llin + (size_t)(b*CDIM + c) * NPIX;
  for (int idx = tid; idx < NPIX; idx += 256) {
    int h = idx / WSZ, w = idx - h*WSZ;
    int i = h >> 1, j = w >> 1, ph = h & 1, pw = w & 1;
    size_t g0 = ((size_t)nb*NPOS + (size_t)(((b*6 + 0)*HH + i)*WH + j))*BSZ + k;
    size_t g5 = ((size_t)nb*NPOS + (size_t)(((b*6 + 5)*HH + i)*WH + j))*BSZ + k;
    P0[idx] = llo[idx];
    P1[idx] = c2q_elem(Xr, Xi, g0, g5, ph, pw);   // lh
  }
  __syncthreads();

  // stage 4: lo = colfilt(ll, G0) + colfilt(lh, G1)
  for (int idx = tid; idx < NPIX; idx += 256) {
    int h = idx / WSZ, w = idx - h*WSZ;
    float acc = 0.f;
#pragma unroll
    for (int t = 0; t < 19; ++t) acc += G0c[t] * P0[symi(h + t - 9, HSZ)*WSZ + w];
#pragma unroll
    for (int t = 0; t < 13; ++t) acc += G1c[t] * P1[symi(h + t - 6, HSZ)*WSZ + w];
    P3[idx] = acc;
  }
  __syncthreads();

  // stage 5: y = rowfilt(lo, G0) + rowfilt(hi, G1) -> out (B, N, C)
  for (int idx = tid; idx < NPIX; idx += 256) {
    int h = idx / WSZ, w = idx - h*WSZ;
    float acc = 0.f;
#pragma unroll
    for (int t = 0; t < 19; ++t) acc += G0c[t] * P3[h*WSZ + symi(w + t - 9, WSZ)];
#pragma unroll
    for (int t = 0; t < 13; ++t) acc += G1c[t] * P2[h*WSZ + symi(w + t - 6, WSZ)];
    out[((size_t)b*NPIX + idx)*CDIM + c] = acc;
  }
}

// ---------------------------------------------------------------------------
extern "C" void kernel_launch(void* const* d_in, const int* in_sizes, int n_in,
                              void* d_out, int out_size, void* d_ws, size_t ws_size,
                              hipStream_t stream)
{
  const float* x    = (const float*)d_in[0];
  const float* w_ll = (const float*)d_in[1];
  const float* w1   = (const float*)d_in[2];
  const float* w2   = (const float*)d_in[3];
  const float* b1   = (const float*)d_in[4];
  const float* b2   = (const float*)d_in[5];
  float* out = (float*)d_out;

  float* ws = (float*)d_ws;
  float* ll = ws;                                       // B*C*NPIX
  float* Xr = ll + (size_t)BATCH * CDIM * NPIX;         // NBLK*NPOS*16
  float* Xi = Xr + (size_t)NBLK * NPOS * BSZ;           // NBLK*NPOS*16

  fwd_dtcwt<<<BATCH*CDIM, 256, 0, stream>>>(x, w_ll, ll, Xr, Xi);
  mix_wmma<<<NTILE/8, 256, 0, stream>>>(w1, w2, b1, b2, Xr, Xi);
  inv_dtcwt<<<BATCH*CDIM, 256, 0, stream>>>(ll, Xr, Xi, out);
}